// LoRAICNN_77300821393779
// MI455X (gfx1250) — compile-verified
//
#include <hip/hip_runtime.h>

#define BN   256
#define DIN  128
#define HN   512
#define RANK 8
#define SCALE 2.0f   // alpha/rank = 16/8
#define AZP  528     // padded LDS row stride for Az (bank-conflict avoidance)

typedef __attribute__((ext_vector_type(2))) float v2f;
typedef __attribute__((ext_vector_type(8))) float v8f;

#define LOG2E 1.44269504088896340736f
#define LN2   0.69314718055994530942f

// softplus in native base-2: sp(v) = max(v,0) + ln2*log2(1 + exp2(-|v|*log2e))
// exp2 arg <= 0 -> e in (0,1], 1+e in (1,2] -> raw v_exp_f32/v_log_f32 are
// safe with no denorm/overflow fixups (unlike __expf/__logf lowering).
__device__ __forceinline__ float softplus_f(float v) {
    float e = __builtin_amdgcn_exp2f(-LOG2E * fabsf(v));
    return fmaxf(v, 0.0f) + LN2 * __builtin_amdgcn_logf(1.0f + e);
}

// LDS byte offset of a __shared__ object (generic -> addrspace(3) -> int)
__device__ __forceinline__ unsigned lds_off(const void* p) {
    return (unsigned)(unsigned long long)
        (__attribute__((address_space(3))) const void*)p;
}

// CDNA5 async DMA: global -> LDS, 16 bytes, tracked by ASYNCcnt.
__device__ __forceinline__ void async_copy_b128(unsigned lds_byte_off,
                                                const void* gptr) {
    asm volatile("global_load_async_to_lds_b128 %0, %1, off"
                 :: "v"(lds_byte_off), "v"(gptr)
                 : "memory");
}

__device__ __forceinline__ void async_wait0() {
    asm volatile("s_wait_asynccnt 0x0" ::: "memory");
}

// ---------------------------------------------------------------------------
// x-path: pre[b][o] = Wx@x + SCALE * Bx@(Ax@x) + bias + bd   (low-rank trick)
// ---------------------------------------------------------------------------
__global__ __launch_bounds__(HN) void xpath_kernel(
    const float* __restrict__ x,    // [B, DIN]
    const float* __restrict__ Wx,   // [H, DIN]
    const float* __restrict__ bias, // [H]
    const float* __restrict__ Ax,   // [B, RANK, DIN]
    const float* __restrict__ Bx,   // [B, H, RANK]
    const float* __restrict__ bd,   // [B, H]
    float* __restrict__ outbuf,     // [B, H]
    int apply_sp)
{
    __shared__ float xs[DIN];
    __shared__ float ts[RANK];
    const int b = blockIdx.x;
    const int t = threadIdx.x;

    if (t < DIN) xs[t] = x[(size_t)b * DIN + t];
    __syncthreads();

    if (t < RANK) {
        const float* arow = Ax + ((size_t)b * RANK + t) * DIN;
        float s = 0.0f;
        #pragma unroll 8
        for (int i = 0; i < DIN; ++i) s = fmaf(arow[i], xs[i], s);
        ts[t] = s;
    }
    __syncthreads();

    const int o = t;
    const float* wrow = Wx + (size_t)o * DIN;
    float s = 0.0f;
    #pragma unroll 8
    for (int i = 0; i < DIN; ++i) s = fmaf(wrow[i], xs[i], s);

    const float* brow = Bx + ((size_t)b * HN + o) * RANK;
    float lora = 0.0f;
    #pragma unroll
    for (int r = 0; r < RANK; ++r) lora = fmaf(brow[r], ts[r], lora);

    float pre = s + SCALE * lora + bias[o] + bd[(size_t)b * HN + o];
    outbuf[(size_t)b * HN + o] = apply_sp ? softplus_f(pre) : pre;
}

// ---------------------------------------------------------------------------
// output-layer x-path (DOUT == 1)
// ---------------------------------------------------------------------------
__global__ __launch_bounds__(DIN) void outx_kernel(
    const float* __restrict__ x,   // [B, DIN]
    const float* __restrict__ Wx,  // [1, DIN]
    const float* __restrict__ ob,  // [1]
    const float* __restrict__ Ax,  // [B, RANK, DIN]
    const float* __restrict__ Bx,  // [B, 1, RANK]
    const float* __restrict__ bd,  // [B, 1]
    float* __restrict__ preout)    // [B]
{
    __shared__ float xs[DIN];
    __shared__ float ts[RANK];
    __shared__ float red[DIN];
    const int b = blockIdx.x;
    const int t = threadIdx.x;

    xs[t] = x[(size_t)b * DIN + t];
    __syncthreads();
    if (t < RANK) {
        const float* arow = Ax + ((size_t)b * RANK + t) * DIN;
        float s = 0.0f;
        #pragma unroll 8
        for (int i = 0; i < DIN; ++i) s = fmaf(arow[i], xs[i], s);
        ts[t] = s;
    }
    red[t] = Wx[t] * xs[t];
    __syncthreads();
    for (int s = DIN / 2; s > 0; s >>= 1) {
        if (t < s) red[t] += red[t + s];
        __syncthreads();
    }
    if (t == 0) {
        float lora = 0.0f;
        #pragma unroll
        for (int r = 0; r < RANK; ++r) lora = fmaf(Bx[(size_t)b * RANK + r], ts[r], lora);
        preout[b] = red[0] + SCALE * lora + ob[0] + bd[b];
    }
}

// ---------------------------------------------------------------------------
// z-path: zout[b][o] = sp( prex[b][o] +
//            sum_i sp(Wz[o][i] + SCALE * (Bz[b] @ Az[b])[o][i]) * zin[b][i] )
//
// One block per sample. Az[b] (8x512, padded rows) + zin row are DMA'd to LDS
// with global_load_async_to_lds_b128. Each of the 8 waves sweeps 4 o-tiles;
// per 16-wide i-tile the rank-8 delta is two chained V_WMMA_F32_16X16X4_F32
// (A = Az^T from LDS, B = Bz^T, i-loop unrolled 2x -> 4 WMMAs in flight).
// The softplus epilogue is staged in passes so the independent v_exp_f32 /
// v_log_f32 ops fill each other's TRANS hazard slots (no v_nop padding).
// ---------------------------------------------------------------------------
__global__ __launch_bounds__(256) void zlayer_kernel(
    const float* __restrict__ zin,  // [B, H]
    const float* __restrict__ Wz,   // [H, H]
    const float* __restrict__ Az,   // [B, RANK, H]
    const float* __restrict__ Bz,   // [B, H, RANK]
    const float* __restrict__ prex, // [B, H]
    float* __restrict__ zout)       // [B, H]
{
    __shared__ __align__(16) float zs[HN];
    __shared__ __align__(16) float azs[RANK * AZP];
    const int b = blockIdx.x;
    const int t = threadIdx.x;

    // --- async staging: zin row (512 f32) + Az[b] (8x512 f32, padded) ------
    {
        const unsigned zs_base = lds_off(&zs[0]);
        const unsigned az_base = lds_off(&azs[0]);
        if (t < HN / 4) {
            async_copy_b128(zs_base + (unsigned)t * 16,
                            zin + (size_t)b * HN + t * 4);
        }
        const float* azg = Az + (size_t)b * RANK * HN;
        #pragma unroll
        for (int j = 0; j < 4; ++j) {
            int n4 = t + 256 * j;       // float4 index, 0..1023
            int n  = n4 * 4;            // element index into Az[b]
            int k  = n >> 9;            // rank row
            int i  = n & (HN - 1);      // column
            async_copy_b128(az_base + (unsigned)(k * AZP + i) * 4, azg + n);
        }
        async_wait0();
    }
    __syncthreads();

    const int lane = t & 31;
    const int wave = t >> 5;
    const int m    = lane & 15;   // A: i_local row; B/C/D: o_local column
    const int half = lane >> 4;   // selects K pair / M+8 rows of C

    // A operand base in LDS: rows K = 2*half (+1, +4, +5), column i0+m
    const float* azp = &azs[(half * 2) * AZP + m];

    for (int ot = 0; ot < 4; ++ot) {
        const int o0 = (ot * 8 + wave) * 16;
        const int o  = o0 + m;

        // B operand (K=rank x N=16), invariant over the i-loop
        const float* bzp = Bz + ((size_t)b * HN + o) * RANK + half * 2;
        v2f blo, bhi;
        blo[0] = bzp[0]; blo[1] = bzp[1];
        bhi[0] = bzp[4]; bhi[1] = bzp[5];

        const float* wbase = Wz + (size_t)o * HN + half * 8;
        float acc0 = 0.0f, acc1 = 0.0f;

        #pragma unroll 2
        for (int i0 = 0; i0 < HN; i0 += 16) {
            v2f alo, ahi;
            alo[0] = azp[i0];             // K = 2*half
            alo[1] = azp[i0 + AZP];       // K = 2*half + 1
            ahi[0] = azp[i0 + 4 * AZP];   // K = 2*half + 4
            ahi[1] = azp[i0 + 5 * AZP];   // K = 2*half + 5

            v8f c = {};
            c = __builtin_amdgcn_wmma_f32_16x16x4_f32(false, alo, false, blo,
                                                      (short)0, c, false, false);
            c = __builtin_amdgcn_wmma_f32_16x16x4_f32(false, ahi, false, bhi,
                                                      (short)0, c, false, false);

            // D[j + 8*half][m] = delta[i0 + j + 8*half][o]
            const float* wrow = wbase + i0;
            __builtin_prefetch(wrow + 64, 0, 3);   // next Wz chunk, keep near
            float4 wa = *(const float4*)(wrow);
            float4 wb = *(const float4*)(wrow + 4);
            float wv[8] = {wa.x, wa.y, wa.z, wa.w, wb.x, wb.y, wb.z, wb.w};
            const float* zp = &zs[i0 + half * 8];

            // staged softplus: keep the 8 transcendental chains independent
            float pre8[8], tr8[8];
            #pragma unroll
            for (int j = 0; j < 8; ++j)
                pre8[j] = fmaf(SCALE, c[j], wv[j]);
            #pragma unroll
            for (int j = 0; j < 8; ++j)
                tr8[j] = __builtin_amdgcn_exp2f(-LOG2E * fabsf(pre8[j]));
            #pragma unroll
            for (int j = 0; j < 8; ++j)
                tr8[j] = __builtin_amdgcn_logf(1.0f + tr8[j]);
            #pragma unroll
            for (int j = 0; j < 8; ++j) {
                float w_eff = fmaf(LN2, tr8[j], fmaxf(pre8[j], 0.0f));
                if (j & 1) acc1 = fmaf(w_eff, zp[j], acc1);
                else       acc0 = fmaf(w_eff, zp[j], acc0);
            }
        }

        float acc = acc0 + acc1;
        // combine the two lane halves (i subsets)
        acc += __shfl_xor(acc, 16, 32);
        if (half == 0) {
            float pre = prex[(size_t)b * HN + o] + acc;
            zout[(size_t)b * HN + o] = softplus_f(pre);
        }
    }
}

// ---------------------------------------------------------------------------
// output z-path (DOUT == 1)
// ---------------------------------------------------------------------------
__global__ __launch_bounds__(256) void outz_kernel(
    const float* __restrict__ z2,     // [B, H]
    const float* __restrict__ Wz,     // [1, H]
    const float* __restrict__ Az,     // [B, RANK, H]
    const float* __restrict__ Bz,     // [B, 1, RANK]
    const float* __restrict__ preout, // [B]
    float* __restrict__ out)          // [B, 1]
{
    __shared__ float red[256];
    const int b = blockIdx.x;
    const int t = threadIdx.x;

    float bz[RANK];
    #pragma unroll
    for (int r = 0; r < RANK; ++r) bz[r] = Bz[(size_t)b * RANK + r];

    const float* azb = Az + (size_t)b * RANK * HN;
    float acc = 0.0f;
    for (int i = t; i < HN; i += 256) {
        float d = 0.0f;
        #pragma unroll
        for (int r = 0; r < RANK; ++r) d = fmaf(bz[r], azb[(size_t)r * HN + i], d);
        float w_eff = softplus_f(fmaf(SCALE, d, Wz[i]));
        acc = fmaf(w_eff, z2[(size_t)b * HN + i], acc);
    }
    red[t] = acc;
    __syncthreads();
    for (int s = 128; s > 0; s >>= 1) {
        if (t < s) red[t] += red[t + s];
        __syncthreads();
    }
    if (t == 0) out[b] = preout[b] + red[0];
}

// ---------------------------------------------------------------------------
extern "C" void kernel_launch(void* const* d_in, const int* in_sizes, int n_in,
                              void* d_out, int out_size, void* d_ws, size_t ws_size,
                              hipStream_t stream)
{
    (void)in_sizes; (void)n_in; (void)out_size; (void)ws_size;
    const float* x      = (const float*)d_in[0];
    const float* Wx0    = (const float*)d_in[1];
    const float* b0     = (const float*)d_in[2];
    const float* Ax0    = (const float*)d_in[3];
    const float* Bx0    = (const float*)d_in[4];
    const float* bd0    = (const float*)d_in[5];
    const float* Wx1    = (const float*)d_in[6];
    const float* b1     = (const float*)d_in[7];
    const float* Ax1    = (const float*)d_in[8];
    const float* Bx1    = (const float*)d_in[9];
    const float* bd1    = (const float*)d_in[10];
    const float* Wx2    = (const float*)d_in[11];
    const float* b2     = (const float*)d_in[12];
    const float* Ax2    = (const float*)d_in[13];
    const float* Bx2    = (const float*)d_in[14];
    const float* bd2    = (const float*)d_in[15];
    const float* Wz1    = (const float*)d_in[16];
    const float* Az1    = (const float*)d_in[17];
    const float* Bz1    = (const float*)d_in[18];
    const float* Wz2    = (const float*)d_in[19];
    const float* Az2    = (const float*)d_in[20];
    const float* Bz2    = (const float*)d_in[21];
    const float* out_Wx = (const float*)d_in[22];
    const float* out_b  = (const float*)d_in[23];
    const float* out_Ax = (const float*)d_in[24];
    const float* out_Bx = (const float*)d_in[25];
    const float* out_bd = (const float*)d_in[26];
    const float* out_Wz = (const float*)d_in[27];
    const float* out_Az = (const float*)d_in[28];
    const float* out_Bz = (const float*)d_in[29];

    float* ws = (float*)d_ws;
    const size_t BH = (size_t)BN * HN;
    float* z0     = ws;
    float* prex1  = ws + 1 * BH;
    float* z1     = ws + 2 * BH;
    float* prex2  = ws + 3 * BH;
    float* z2     = ws + 4 * BH;
    float* preout = ws + 5 * BH;

    // x-paths (low-rank, independent of z)
    xpath_kernel<<<BN, HN, 0, stream>>>(x, Wx0, b0, Ax0, Bx0, bd0, z0, 1);
    xpath_kernel<<<BN, HN, 0, stream>>>(x, Wx1, b1, Ax1, Bx1, bd1, prex1, 0);
    xpath_kernel<<<BN, HN, 0, stream>>>(x, Wx2, b2, Ax2, Bx2, bd2, prex2, 0);
    outx_kernel<<<BN, DIN, 0, stream>>>(x, out_Wx, out_b, out_Ax, out_Bx, out_bd, preout);

    // z-paths (async LDS staging + WMMA delta + softplus'd per-sample weights)
    zlayer_kernel<<<BN, 256, 0, stream>>>(z0, Wz1, Az1, Bz1, prex1, z1);
    zlayer_kernel<<<BN, 256, 0, stream>>>(z1, Wz2, Az2, Bz2, prex2, z2);

    outz_kernel<<<BN, 256, 0, stream>>>(z2, out_Wz, out_Az, out_Bz, preout, (float*)d_out);
}